// MaskGenerator_43946105372909
// MI455X (gfx1250) — compile-verified
//
#include <hip/hip_runtime.h>

// Problem constants (from reference): B=2, H=8, S=2048, SPARSITY=0.8, LOCAL_WINDOW=64
#define B_     2
#define H_     8
#define S_     2048
#define K_     409          // max(1, int(S * (1 - SPARSITY)))
#define HALF_  32           // LOCAL_WINDOW / 2
#define NT     256          // threads per block (8 wave32s)
#define NW     8            // waves per block
#define EPT    8            // elements per thread (NT * EPT == S_)

typedef __attribute__((ext_vector_type(4))) unsigned int v4u;
typedef __attribute__((ext_vector_type(8))) int          v8i;
typedef __attribute__((ext_vector_type(4))) int          v4i;

// Order-preserving float -> uint mapping (larger float => larger key).
__device__ __forceinline__ unsigned f2ord(float f) {
  unsigned u = __float_as_uint(f);
  return (u & 0x80000000u) ? ~u : (u | 0x80000000u);
}

// Intra-wave inclusive SUFFIX sum (wave32, no barriers; lane ops).
__device__ __forceinline__ unsigned wave_suffix_incl(unsigned v, unsigned lane) {
#pragma unroll
  for (int off = 1; off < 32; off <<= 1) {
    unsigned o = __shfl_down(v, off, 32);
    v += (lane + (unsigned)off < 32u) ? o : 0u;
  }
  return v;
}

// Intra-wave inclusive PREFIX sum (wave32, no barriers; lane ops).
__device__ __forceinline__ unsigned wave_prefix_incl(unsigned v, unsigned lane) {
#pragma unroll
  for (int off = 1; off < 32; off <<= 1) {
    unsigned o = __shfl_up(v, off, 32);
    v += (lane >= (unsigned)off) ? o : 0u;
  }
  return v;
}

// Exact top-K_ selection over a 2048-key row held as key[EPT] per thread
// (thread t owns consecutive columns t*EPT .. t*EPT+7). Matches jax.lax.top_k
// tie-breaking (lower index wins) via threshold + stable rank among equals.
// 4 radix passes (8-bit digits), 256-bin LDS histogram; scans are hierarchical:
// shfl-based intra-wave + 8-entry cross-wave combine (~4 barriers/round).
__device__ __forceinline__ void topk_select(const unsigned* key, unsigned tid,
                                            unsigned* s_hist, unsigned* s_wsum,
                                            unsigned* s_bc, unsigned char* sel) {
  const unsigned lane = tid & 31u;
  const unsigned wid  = tid >> 5;
  unsigned prefix = 0u, pmask = 0u;
  unsigned remaining = (unsigned)K_;

#pragma unroll 1
  for (int round = 0; round < 4; ++round) {
    const int shift = 24 - 8 * round;
    s_hist[tid] = 0u;
    __syncthreads();
#pragma unroll
    for (int e = 0; e < EPT; ++e) {
      if ((key[e] & pmask) == prefix)
        atomicAdd(&s_hist[(key[e] >> shift) & 0xFFu], 1u);
    }
    __syncthreads();
    // Hierarchical inclusive suffix sum over the 256 bins.
    unsigned v   = s_hist[tid];
    unsigned suf = wave_suffix_incl(v, lane);       // bins [tid .. end-of-wave]
    if (lane == 0) s_wsum[wid] = suf;               // wave totals
    __syncthreads();
    unsigned hi = 0u;
#pragma unroll
    for (unsigned w = 0; w < NW; ++w) hi += (w > wid) ? s_wsum[w] : 0u;
    const unsigned mine  = suf + hi;                // sum(hist[tid..255])
    const unsigned above = mine - v;                // sum(hist[tid+1..255])
    if (mine >= remaining && above < remaining) {
      s_bc[0] = tid;                                // selected digit
      s_bc[1] = remaining - above;                  // remaining in this class
    }
    __syncthreads();
    prefix |= s_bc[0] << shift;
    pmask  |= 0xFFu << shift;
    remaining = s_bc[1];
  }

  const unsigned T = prefix;    // exact threshold key; take `remaining` equals
  // Stable rank among keys == T (ascending column index): hierarchical prefix.
  unsigned ceq = 0u;
#pragma unroll
  for (int e = 0; e < EPT; ++e) ceq += (key[e] == T) ? 1u : 0u;
  unsigned pre = wave_prefix_incl(ceq, lane);       // inclusive within wave
  if (lane == 31) s_wsum[wid] = pre;                // wave totals
  __syncthreads();
  unsigned lo = 0u;
#pragma unroll
  for (unsigned w = 0; w < NW; ++w) lo += (w < wid) ? s_wsum[w] : 0u;
  unsigned rank = lo + pre - ceq;                   // exclusive rank of equals
#pragma unroll
  for (int e = 0; e < EPT; ++e) {
    unsigned k = key[e];
    unsigned char s;
    if (k > T)        s = 1;
    else if (k == T) { s = (rank < remaining) ? 1 : 0; ++rank; }
    else              s = 0;
    sel[e] = s;
  }
}

// Kernel 1: per (b,i) row -> head-mean importance -> top-k -> bit-packed gmask.
__global__ __launch_bounds__(NT)
void gmask_kernel(const float* __restrict__ scores, unsigned char* __restrict__ gbytes) {
  __shared__ unsigned s_hist[NT];
  __shared__ unsigned s_wsum[NW];
  __shared__ unsigned s_bc[2];

  const unsigned tid = threadIdx.x;
  const unsigned row = blockIdx.x;          // b*S + i
  const unsigned b   = row / S_;
  const unsigned i   = row - b * S_;
  const unsigned j0  = tid * EPT;

  float acc[EPT];
#pragma unroll
  for (int e = 0; e < EPT; ++e) acc[e] = 0.f;

#pragma unroll
  for (int h = 0; h < H_; ++h) {
    const float4* p = reinterpret_cast<const float4*>(
        scores + (((size_t)(b * H_ + h) * S_ + i) * S_ + j0));
    float4 a0 = p[0];
    float4 a1 = p[1];
    acc[0] += a0.x; acc[1] += a0.y; acc[2] += a0.z; acc[3] += a0.w;
    acc[4] += a1.x; acc[5] += a1.y; acc[6] += a1.z; acc[7] += a1.w;
  }

  unsigned key[EPT];
#pragma unroll
  for (int e = 0; e < EPT; ++e) key[e] = f2ord(acc[e] * 0.125f);

  unsigned char sel[EPT];
  topk_select(key, tid, s_hist, s_wsum, s_bc, sel);

  unsigned byte = 0;
#pragma unroll
  for (int e = 0; e < EPT; ++e) byte |= ((unsigned)sel[e]) << e;
  gbytes[(size_t)row * (S_ / 8) + tid] = (unsigned char)byte;
}

// Kernel 2: per (b,h,i) row -> combined mask -> masked top-k -> binary output.
// Score row is staged into LDS via the Tensor Data Mover; the DMA overlaps the
// rand/gmask global loads before the tensorcnt wait.
__global__ __launch_bounds__(NT)
void final_kernel(const float* __restrict__ scores, const float* __restrict__ rand_u,
                  const unsigned char* __restrict__ gbytes, float* __restrict__ out) {
  __shared__ float    s_row[S_];
  __shared__ unsigned s_hist[NT];
  __shared__ unsigned s_wsum[NW];
  __shared__ unsigned s_bc[2];

  const unsigned tid = threadIdx.x;
  const unsigned row = blockIdx.x;          // (b*H + h)*S + i
  const unsigned i   = row % S_;
  const unsigned bh  = row / S_;
  const unsigned b   = bh / H_;
  const unsigned j0  = tid * EPT;
  const size_t rowOff = (size_t)row * S_;

  // --- TDM: DMA the 2048-float score row into LDS (wave 0 issues) ---
  if (tid < 32) {
    unsigned long long ga = (unsigned long long)(uintptr_t)(scores + rowOff);
    unsigned ldsa = (unsigned)(uintptr_t)(void*)s_row;  // LDS byte offset (low 32b)
    // D# group 0: count=1 | lds_addr | global_addr[56:0] | type=2
    v4u g0 = { 1u,
               ldsa,
               (unsigned)(ga & 0xFFFFFFFFull),
               (unsigned)((ga >> 32) & 0x1FFFFFFull) | (2u << 30) };
    // D# group 1: data_size=4B; tensor_dim0=2048, tensor_dim1=1;
    //             tile_dim0=2048, tile_dim1=1; tensor_dim0_stride=2048
    v8i g1 = { (int)(2u << 16),       // data_size = 2 (4 bytes)
               (int)(2048u << 16),    // tensor_dim0[15:0]
               (int)(1u << 16),       // tensor_dim0 hi16 = 0 | tensor_dim1[15:0] = 1
               (int)(2048u << 16),    // tensor_dim1 hi16 = 0 | tile_dim0 = 2048
               1,                     // tile_dim1 = 1, tile_dim2 = 0
               2048,                  // tensor_dim0_stride[31:0]
               0, 0 };                // stride hi, tensor_dim1_stride = 0
    v4i z4 = { 0, 0, 0, 0 };
    v8i z8 = { 0, 0, 0, 0, 0, 0, 0, 0 };
    __builtin_amdgcn_tensor_load_to_lds(g0, g1, z4, z4, z8, 0);
  }

  // Overlap the DMA with the rand row + gmask byte loads.
  const float4* rp = reinterpret_cast<const float4*>(rand_u + rowOff + j0);
  float4 rv0 = rp[0];
  float4 rv1 = rp[1];
  const unsigned gb = (unsigned)gbytes[((size_t)b * S_ + i) * (S_ / 8) + tid];

  if (tid < 32) __builtin_amdgcn_s_wait_tensorcnt(0);
  __syncthreads();

  float4 sv0 = reinterpret_cast<const float4*>(s_row + j0)[0];
  float4 sv1 = reinterpret_cast<const float4*>(s_row + j0)[1];

  float sval[EPT] = { sv0.x, sv0.y, sv0.z, sv0.w, sv1.x, sv1.y, sv1.z, sv1.w };
  float rval[EPT] = { rv0.x, rv0.y, rv0.z, rv0.w, rv1.x, rv1.y, rv1.z, rv1.w };

  unsigned key[EPT];
#pragma unroll
  for (int e = 0; e < EPT; ++e) {
    const int j = (int)(j0 + e);
    const bool loc  = (j >= (int)i - HALF_) && (j < (int)i + HALF_);
    const bool gm   = (gb >> e) & 1u;
    const bool rnd  = rval[e] < 0.2f;      // float32(1.0 - SPARSITY) == 0.2f
    const bool comb = loc | gm | rnd;
    key[e] = f2ord(comb ? sval[e] : 0.0f);
  }

  unsigned char sel[EPT];
  topk_select(key, tid, s_hist, s_wsum, s_bc, sel);

  float4 o0 = make_float4(sel[0] ? 1.f : 0.f, sel[1] ? 1.f : 0.f,
                          sel[2] ? 1.f : 0.f, sel[3] ? 1.f : 0.f);
  float4 o1 = make_float4(sel[4] ? 1.f : 0.f, sel[5] ? 1.f : 0.f,
                          sel[6] ? 1.f : 0.f, sel[7] ? 1.f : 0.f);
  float4* op = reinterpret_cast<float4*>(out + rowOff + j0);
  op[0] = o0;
  op[1] = o1;
}

extern "C" void kernel_launch(void* const* d_in, const int* in_sizes, int n_in,
                              void* d_out, int out_size, void* d_ws, size_t ws_size,
                              hipStream_t stream) {
  const float* scores = (const float*)d_in[0];   // (B,H,S,S) f32
  const float* rand_u = (const float*)d_in[1];   // (B,H,S,S) f32
  float* out = (float*)d_out;                    // (B,H,S,S) f32
  unsigned char* gbytes = (unsigned char*)d_ws;  // B*S*(S/8) = 1 MiB bit-packed gmask

  gmask_kernel<<<dim3(B_ * S_), dim3(NT), 0, stream>>>(scores, gbytes);
  final_kernel<<<dim3(B_ * H_ * S_), dim3(NT), 0, stream>>>(scores, rand_u, gbytes, out);
}